// EIGLayer_69346541962061
// MI455X (gfx1250) — compile-verified
//
#include <hip/hip_runtime.h>
#include <hip/hip_bf16.h>

typedef __attribute__((ext_vector_type(2))) float v2f;
typedef __attribute__((ext_vector_type(8))) float v8f;

#define NN 40000
#define EE 640000
#define DD 128
#define PL ((long)NN * DD)           // one aggregate plane, in floats
#define AVG_D_LOG 2.772588722239781f // log(16)
#define KC_PRE 68                    // 272 / 4 k-chunks
#define KC_POST 608                  // 2432 / 4 k-chunks

// ws layout (floats):
//   P[6*PL]      aggregate planes: 0 mean, 1 max, 2 min, 3 std, 4 dir_av, 5 dir_dx
//   deg[NN], denom[NN]
//   WpackP[KC_PRE*512], WpackQ[KC_POST*512]   (WMMA-fragment-ordered weights)

__global__ __launch_bounds__(256) void init_ws(float* P, float* deg, float* denom) {
    long i = (long)blockIdx.x * 256 + threadIdx.x;
    if (i < 6 * PL) {
        long p = i / PL;
        float v = 0.0f;
        if (p == 1) v = -__builtin_inff();
        else if (p == 2) v = __builtin_inff();
        P[i] = v;
    } else if (i < 6 * PL + NN) {
        deg[i - 6 * PL] = 0.0f;
    } else if (i < 6 * PL + 2 * NN) {
        denom[i - 6 * PL - NN] = 0.0f;
    }
}

// Repack row-major W[K x 128] into per-fragment order:
// pair index ((kc*8 + nt)*32 + lane) holds (W[k][col], W[k+1][col]),
// k = kc*4 + 2*(lane>>4), col = nt*16 + (lane&15)  — one b64 per fragment.
__global__ __launch_bounds__(256) void pack_w(const float* __restrict__ W,
                                              float* __restrict__ out, int nchunks) {
    int tid = blockIdx.x * 256 + threadIdx.x;
    if (tid >= nchunks * 256) return;
    int kc = tid >> 8;
    int rem = tid & 255;
    int nt = rem >> 5;
    int lane = rem & 31;
    int k = kc * 4 + ((lane >> 4) << 1);
    int col = nt * 16 + (lane & 15);
    out[(long)tid * 2]     = W[(long)k * DD + col];
    out[(long)tid * 2 + 1] = W[(long)(k + 1) * DD + col];
}

__global__ __launch_bounds__(256) void deg_denom_kernel(const float* __restrict__ eig,
                                                        const int* __restrict__ src,
                                                        const int* __restrict__ dst,
                                                        float* __restrict__ deg,
                                                        float* __restrict__ denom) {
    int e = blockIdx.x * 256 + threadIdx.x;
    if (e >= EE) return;
    int s = src[e], t = dst[e];
    float d = eig[2 * s + 1] - eig[2 * t + 1];
    atomicAdd(denom + t, fabsf(d));
    atomicAdd(deg + t, 1.0f);
}

// Fused pretrans GEMM (f32 WMMA, 32 edges x 128 out per wave, K=272) + atomic scatter.
__global__ __launch_bounds__(128) void pretrans_agg(const float* __restrict__ h,
                                                    const float* __restrict__ ef,
                                                    const float* __restrict__ eig,
                                                    const int* __restrict__ src,
                                                    const int* __restrict__ dst,
                                                    const v2f* __restrict__ Wpack,
                                                    const float* __restrict__ bp,
                                                    const float* __restrict__ denomArr,
                                                    float* __restrict__ P) {
    const int lane = threadIdx.x & 31;
    const int wave = threadIdx.x >> 5;
    const int e0 = (blockIdx.x * 4 + wave) * 32;

    __shared__ int   sDst[4][32];
    __shared__ float sWav[4][32];
    __shared__ float sWdx[4][32];

    {
        int e = e0 + lane;
        int s = src[e], t = dst[e];
        float d = eig[2 * s + 1] - eig[2 * t + 1];
        float den = denomArr[t] + 1e-8f;
        sDst[wave][lane] = t;
        sWav[wave][lane] = fabsf(d) / den;
        sWdx[wave][lane] = d / den;
    }
    __syncthreads();

    const int row  = lane & 15;        // A-matrix row within tile
    const int koff = (lane >> 4) * 2;  // K sub-offset per ISA A layout
    const int col0 = lane & 15;
    const int er0  = e0 + row;
    const int er1  = e0 + 16 + row;
    const long sO0 = (long)src[er0] * DD, tO0 = (long)dst[er0] * DD, fO0 = (long)er0 * 16;
    const long sO1 = (long)src[er1] * DD, tO1 = (long)dst[er1] * DD, fO1 = (long)er1 * 16;

    v8f acc0[8], acc1[8];
#pragma unroll
    for (int i = 0; i < 8; ++i) { acc0[i] = (v8f)(0.0f); acc1[i] = (v8f)(0.0f); }

    auto step = [&](v2f a0, v2f a1, int kc) {
        const v2f* wkc = Wpack + (long)kc * 256 + lane;
#pragma unroll
        for (int nt = 0; nt < 8; ++nt) {
            v2f b = wkc[nt * 32];
            acc0[nt] = __builtin_amdgcn_wmma_f32_16x16x4_f32(
                false, a0, false, b, (short)0, acc0[nt], false, false);
            acc1[nt] = __builtin_amdgcn_wmma_f32_16x16x4_f32(
                false, a1, false, b, (short)0, acc1[nt], false, false);
        }
    };

    // K segments branch-free: [0,128)=h[src], [128,256)=h[dst], [256,272)=e
    for (int kc = 0; kc < 32; ++kc) {
        int k = kc * 4 + koff;
        step(*(const v2f*)(h + sO0 + k), *(const v2f*)(h + sO1 + k), kc);
    }
    for (int kc = 0; kc < 32; ++kc) {
        int k = kc * 4 + koff;
        step(*(const v2f*)(h + tO0 + k), *(const v2f*)(h + tO1 + k), kc + 32);
    }
    for (int kc = 0; kc < 4; ++kc) {
        int k = kc * 4 + koff;
        step(*(const v2f*)(ef + fO0 + k), *(const v2f*)(ef + fO1 + k), kc + 64);
    }

    float* Psum = P;
    float* Pmax = P + PL;
    float* Pmin = P + 2 * PL;
    float* Psq  = P + 3 * PL;
    float* Pav  = P + 4 * PL;
    float* Pdx  = P + 5 * PL;

    const int rbase = (lane >> 4) * 8;
#pragma unroll
    for (int nt = 0; nt < 8; ++nt) {
        int c = nt * 16 + col0;
        float bv = bp[c];
#pragma unroll
        for (int r = 0; r < 8; ++r) {
            int rr = rbase + r;
            {
                float m = acc0[nt][r] + bv;
                int t = sDst[wave][rr];
                long o = (long)t * DD + c;
                atomicAdd(Psum + o, m);
                atomicAdd(Psq  + o, m * m);
                atomicMax(Pmax + o, m);
                atomicMin(Pmin + o, m);
                atomicAdd(Pav  + o, m * sWav[wave][rr]);
                atomicAdd(Pdx  + o, m * sWdx[wave][rr]);
            }
            {
                float m = acc1[nt][r] + bv;
                int t = sDst[wave][16 + rr];
                long o = (long)t * DD + c;
                atomicAdd(Psum + o, m);
                atomicAdd(Psq  + o, m * m);
                atomicMax(Pmax + o, m);
                atomicMin(Pmin + o, m);
                atomicAdd(Pav  + o, m * sWav[wave][16 + rr]);
                atomicAdd(Pdx  + o, m * sWdx[wave][16 + rr]);
            }
        }
    }
}

__global__ __launch_bounds__(256) void finalize_agg(const float* __restrict__ h,
                                                    const float* __restrict__ degA,
                                                    float* __restrict__ P) {
    long idx = (long)blockIdx.x * 256 + threadIdx.x;
    if (idx >= PL) return;
    int n = (int)(idx >> 7);
    float deg = degA[n];
    bool has = deg > 0.0f;
    float degc = fmaxf(deg, 1.0f);

    float sum = P[idx];
    float mean = sum / degc;
    float sq = P[3 * PL + idx];
    float var = sq / degc - mean * mean;
    float stdv = sqrtf(fmaxf(var, 0.0f) + 1e-5f);

    P[idx]          = has ? mean : 0.0f;
    P[PL + idx]     = has ? P[PL + idx] : 0.0f;
    P[2 * PL + idx] = has ? P[2 * PL + idx] : 0.0f;
    P[3 * PL + idx] = has ? stdv : 0.0f;
    P[4 * PL + idx] = has ? P[4 * PL + idx] : 0.0f;
    float dx = P[5 * PL + idx] - h[idx];
    P[5 * PL + idx] = has ? dx : 0.0f;
}

// Fused posttrans GEMM (f32 WMMA, 32 nodes x 128 out per wave, K=2432).
__global__ __launch_bounds__(64) void posttrans(const float* __restrict__ h,
                                                const float* __restrict__ snorm,
                                                const float* __restrict__ degA,
                                                const v2f* __restrict__ Wpack,
                                                const float* __restrict__ bq,
                                                const float* __restrict__ P,
                                                float* __restrict__ out) {
    const int lane = threadIdx.x & 31;
    const int wave = threadIdx.x >> 5;
    const int n0 = (blockIdx.x * 2 + wave) * 32;

    const int row  = lane & 15;
    const int koff = (lane >> 4) * 2;
    const int col0 = lane & 15;
    const int nA = n0 + row;
    const int nB = n0 + 16 + row;
    const long hA = (long)nA * DD;
    const long hB = (long)nB * DD;

    float logdA = logf(fmaxf(degA[nA], 1.0f) + 1.0f);
    float logdB = logf(fmaxf(degA[nB], 1.0f) + 1.0f);
    const float ampA = logdA / AVG_D_LOG, attA = AVG_D_LOG / logdA;
    const float ampB = logdB / AVG_D_LOG, attB = AVG_D_LOG / logdB;

    v8f acc0[8], acc1[8];
#pragma unroll
    for (int i = 0; i < 8; ++i) { acc0[i] = (v8f)(0.0f); acc1[i] = (v8f)(0.0f); }

    auto step = [&](v2f a0, v2f a1, int kc) {
        const v2f* wkc = Wpack + (long)kc * 256 + lane;
#pragma unroll
        for (int nt = 0; nt < 8; ++nt) {
            v2f b = wkc[nt * 32];
            acc0[nt] = __builtin_amdgcn_wmma_f32_16x16x4_f32(
                false, a0, false, b, (short)0, acc0[nt], false, false);
            acc1[nt] = __builtin_amdgcn_wmma_f32_16x16x4_f32(
                false, a1, false, b, (short)0, acc1[nt], false, false);
        }
    };

    // Segment 1: k in [0,128) -> h rows
    for (int kc = 0; kc < 32; ++kc) {
        int k = kc * 4 + koff;
        step(*(const v2f*)(h + hA + k), *(const v2f*)(h + hB + k), kc);
    }
    // Segments 2..19: k in [128,2432) -> agg planes x {1, amp, att}
    for (int sc = 0; sc < 3; ++sc) {
        float sA = (sc == 0) ? 1.0f : ((sc == 1) ? ampA : attA);
        float sB = (sc == 0) ? 1.0f : ((sc == 1) ? ampB : attB);
        for (int plane = 0; plane < 6; ++plane) {
            const float* pA = P + (long)plane * PL + hA;
            const float* pB = P + (long)plane * PL + hB;
            const int kcbase = 32 + (sc * 6 + plane) * 32;
            for (int kc = 0; kc < 32; ++kc) {
                int c = kc * 4 + koff;
                v2f a0 = *(const v2f*)(pA + c);
                v2f a1 = *(const v2f*)(pB + c);
                a0 = a0 * sA;
                a1 = a1 * sB;
                step(a0, a1, kcbase + kc);
            }
        }
    }

    const int rbase = (lane >> 4) * 8;
#pragma unroll
    for (int nt = 0; nt < 8; ++nt) {
        int c = nt * 16 + col0;
        float bb = bq[c];
#pragma unroll
        for (int r = 0; r < 8; ++r) {
            int rr = rbase + r;
            {
                int nn = n0 + rr;
                long o = (long)nn * DD + c;
                float v = (acc0[nt][r] + bb) * snorm[nn];
                out[o] = h[o] + fmaxf(v, 0.0f);
            }
            {
                int nn = n0 + 16 + rr;
                long o = (long)nn * DD + c;
                float v = (acc1[nt][r] + bb) * snorm[nn];
                out[o] = h[o] + fmaxf(v, 0.0f);
            }
        }
    }
}

extern "C" void kernel_launch(void* const* d_in, const int* in_sizes, int n_in,
                              void* d_out, int out_size, void* d_ws, size_t ws_size,
                              hipStream_t stream) {
    const float* h     = (const float*)d_in[0];
    const float* e     = (const float*)d_in[1];
    const float* eig   = (const float*)d_in[2];
    const float* snorm = (const float*)d_in[3];
    const int*   src   = (const int*)d_in[4];
    const int*   dst   = (const int*)d_in[5];
    const float* Wp    = (const float*)d_in[6];
    const float* bp    = (const float*)d_in[7];
    const float* Wq    = (const float*)d_in[8];
    const float* bq    = (const float*)d_in[9];

    float* P      = (float*)d_ws;              // 6 planes of [N,128]
    float* deg    = P + 6 * PL;                // [N]
    float* denom  = deg + NN;                  // [N]
    float* WpackP = denom + NN;                // [KC_PRE * 512]
    float* WpackQ = WpackP + KC_PRE * 512;     // [KC_POST * 512]
    float* out    = (float*)d_out;

    long initTotal = 6 * PL + 2 * NN;
    int initBlocks = (int)((initTotal + 255) / 256);
    init_ws<<<initBlocks, 256, 0, stream>>>(P, deg, denom);

    pack_w<<<KC_PRE, 256, 0, stream>>>(Wp, WpackP, KC_PRE);
    pack_w<<<KC_POST, 256, 0, stream>>>(Wq, WpackQ, KC_POST);

    deg_denom_kernel<<<EE / 256, 256, 0, stream>>>(eig, src, dst, deg, denom);

    pretrans_agg<<<EE / 128, 128, 0, stream>>>(h, e, eig, src, dst,
                                               (const v2f*)WpackP, bp, denom, P);

    finalize_agg<<<(int)((PL + 255) / 256), 256, 0, stream>>>(h, deg, P);

    posttrans<<<NN / 64, 64, 0, stream>>>(h, snorm, deg,
                                          (const v2f*)WpackQ, bq, P, out);
}